// LCN_31490700215130
// MI455X (gfx1250) — compile-verified
//
#include <hip/hip_runtime.h>

// ---------------- architecture constants (match reference) ----------------
#define BB    64          // batch
#define INP   128         // input H/W
#define SS_   9           // V1 kernel
#define DIMS  120         // V1 output grid
#define SN    8           // orientations
#define PS    4           // phases
#define CC    32          // V1 channels = SN*PS
#define K1    81          // ss*ss
#define K1P   84          // K1 padded to multiple of 4 (TDM writes zeros via OOB)
#define LDA1  86          // LDS row stride for patches: even (8B align), bank rot 22

#define V4K   8
#define V4S   4
#define V4D   29
#define ORI   16
#define K2    64          // per-orientation chunk of V4 K (v4k*v4k)
#define LDA2  66          // LDS row stride: 64 + 2 pad dwords (TDM pad feature)
#define V4FLAT (ORI * V4D * V4D)   // 13456

typedef __attribute__((ext_vector_type(2))) float    v2f;
typedef __attribute__((ext_vector_type(8))) float    v8f;
typedef __attribute__((ext_vector_type(4))) unsigned tdm4;   // D# group0/2/3
typedef __attribute__((ext_vector_type(8))) unsigned tdm8;   // D# group1

// Issue one Tensor-Data-Mover load (Global -> LDS). Descriptor groups live in
// SGPRs; EXEC is ignored by the TDM so the caller must branch so only one wave
// reaches this. Tracked on TENSORcnt.
__device__ __forceinline__ void tdm_load(tdm4 g0, tdm8 g1, tdm4 g2, tdm4 g3) {
    asm volatile("tensor_load_to_lds %0, %1, %2, %3"
                 :: "s"(g0), "s"(g1), "s"(g2), "s"(g3)
                 : "memory");
}

// =====================================================================
// Kernel 1: V1 locally-connected layer + ReLU + phase pooling.
// One workgroup per (i,j) of the 120x120 grid.
// GEMM per location: [64 x 81] (patches) @ [81 x 32] (w^T) -> [64 x 32]
// 8 waves, each owns one 16x16 tile; K padded 81 -> 84 -> 21 wmma_f32_16x16x4.
// Weights (the 149MB HBM stream) arrive via TDM: 2D tile 32 rows x 84,
// tensor_dim0=81 so the 3 pad elements are OOB-zeros.
// =====================================================================
__global__ __launch_bounds__(256) void lcn_v1_kernel(
    const float* __restrict__ x,      // [64,1,128,128]
    const float* __restrict__ sw,     // [32,120,120,81]
    float* __restrict__ cplx)         // [64,8,120,120]
{
    __shared__ float a_lds[BB * LDA1];   // patches  64 x 84 (stride 86)
    __shared__ float w_lds[CC * K1P];    // weights  32 x 84 (TDM-filled, stride 84)
    __shared__ float o_lds[BB * CC];     // relu(v1) 64 x 32

    const int blk = blockIdx.x;
    const int i = blk / DIMS;
    const int j = blk - i * DIMS;
    const int tid = threadIdx.x;

    // ---- wave 0: kick off the weight tile DMA (overlaps patch staging) ----
    if (tid < 32) {
        const unsigned ldsw = (unsigned)(size_t)(void*)w_lds;
        const unsigned long long ga =
            (unsigned long long)(size_t)(const void*)(sw + ((size_t)i * DIMS + j) * K1);
        tdm4 g0;
        g0[0] = 1u;                                             // count=1 (valid)
        g0[1] = ldsw;                                           // lds_addr
        g0[2] = (unsigned)ga;                                   // global_addr lo
        g0[3] = (unsigned)((ga >> 32) & 0x01FFFFFFull) | (2u << 30); // hi + type=2
        tdm8 g1;
        g1[0] = 2u << 16;                                       // data_size = 4B
        g1[1] = (unsigned)K1 << 16;                             // tensor_dim0 = 81
        g1[2] = (unsigned)CC << 16;                             // tensor_dim1 = 32
        g1[3] = (unsigned)K1P << 16;                            // tile_dim0  = 84
        g1[4] = (unsigned)CC;                                   // tile_dim1  = 32
        g1[5] = (unsigned)(DIMS * DIMS * K1);                   // dim0_stride (c-stride)
        g1[6] = 0u; g1[7] = 0u;
        tdm4 gz; gz[0] = 0u; gz[1] = 0u; gz[2] = 0u; gz[3] = 0u;
        tdm_load(g0, g1, gz, gz);
    }

    // ---- stage A (input patches, L2-resident) into LDS; zero K padding ----
    for (int e = tid; e < BB * K1P; e += 256) {
        const int b = e / K1P, k = e - b * K1P;
        float v = 0.f;
        if (k < K1) {
            const int kh = k / SS_, kw = k - kh * SS_;
            v = x[(b * INP + (i + kh)) * INP + (j + kw)];
        }
        a_lds[b * LDA1 + k] = v;
    }
    if (tid < 32) __builtin_amdgcn_s_wait_tensorcnt(0);
    __syncthreads();

    const int lane = tid & 31;
    const int wave = tid >> 5;
    const int mt   = wave >> 1;      // M tile 0..3
    const int nt   = wave & 1;       // N tile 0..1
    const int half = lane >> 4;      // 0: K=0,1 | 1: K=2,3  (ISA A/B layout)
    const int l16  = lane & 15;
    const int koff = half * 2;

    const float* ap = &a_lds[(mt * 16 + l16) * LDA1 + koff];  // A row = batch
    const float* bp = &w_lds[(nt * 16 + l16) * K1P + koff];   // B col = channel

    v8f acc = {0.f, 0.f, 0.f, 0.f, 0.f, 0.f, 0.f, 0.f};
#pragma unroll
    for (int kk = 0; kk < K1P / 4; ++kk) {                    // 21 steps
        v2f a = *(const v2f*)(ap + kk * 4);
        v2f b = *(const v2f*)(bp + kk * 4);
        acc = __builtin_amdgcn_wmma_f32_16x16x4_f32(
            false, a, false, b, (short)0, acc, false, false);
    }

    // D layout: VGPR v -> M = v + half*8, N = l16  (16x16 f32 C/D tile)
    const int ncol = nt * 16 + l16;
#pragma unroll
    for (int v = 0; v < 8; ++v) {
        const int m = mt * 16 + half * 8 + v;
        o_lds[m * CC + ncol] = fmaxf(acc[v], 0.f);            // ReLU
    }
    __syncthreads();

    // ---- complex-cell pooling: mean over 4 phases -> [64 x 8] ----
    for (int e = tid; e < BB * SN; e += 256) {
        const int b = e >> 3, s = e & 7;
        const float* r = &o_lds[b * CC + s * PS];
        const float m = (r[0] + r[1] + r[2] + r[3]) * 0.25f;
        cplx[((b * SN + s) * DIMS + i) * DIMS + j] = m;
    }
}

// =====================================================================
// Kernel 2: V4 locally-connected pooling.
// One workgroup per (i,j) of the 29x29 grid.
// GEMM per location: [64 x 512] @ [512 x 16] -> [64 x 16],
// K chunked by orientation s (8 chunks of 64 -> 16 wmma steps each).
// 4 waves = 4 M tiles; single N tile. Both tiles arrive via TDM each chunk:
//   A: 3D tile (8 w, 8 h, 64 batches) from cplx
//   W: 2D tile (64 k, 16 o) from v4 weights
// LDS pad feature (interval 64 dw, amount 2 dw) yields stride-66 rows.
// =====================================================================
__global__ __launch_bounds__(128) void lcn_v4_kernel(
    const float* __restrict__ cplx,   // [64,8,120,120]
    const float* __restrict__ vw,     // [16,8,29,29,64]
    float* __restrict__ v4)           // [64, 13456] = [64,16,29,29]
{
    __shared__ float a_lds[BB * LDA2];    // patches 64 x 64 (stride 66)
    __shared__ float w_lds[ORI * LDA2];   // weights 16 x 64 (stride 66)

    const int blk = blockIdx.x;
    const int i = blk / V4D;
    const int j = blk - i * V4D;
    const int tid = threadIdx.x;
    const int lane = tid & 31;
    const int mt   = tid >> 5;            // M tile 0..3
    const int half = lane >> 4;
    const int l16  = lane & 15;
    const int koff = half * 2;

    const float* ap = &a_lds[(mt * 16 + l16) * LDA2 + koff];
    const float* bp = &w_lds[l16 * LDA2 + koff];

    // pad_enable | pad_interval=5 (64 dwords) | pad_amount=1 (2 dwords), 4B data
    const unsigned padctl = (2u << 16) | (1u << 20) | (5u << 22) | (1u << 25);

    v8f acc = {0.f, 0.f, 0.f, 0.f, 0.f, 0.f, 0.f, 0.f};

    for (int s = 0; s < SN; ++s) {
        __syncthreads();   // all waves done reading LDS before DMA overwrite
        if (tid < 32) {
            const unsigned lda = (unsigned)(size_t)(void*)a_lds;
            const unsigned ldw = (unsigned)(size_t)(void*)w_lds;
            const unsigned long long gaA = (unsigned long long)(size_t)(const void*)
                (cplx + ((size_t)s * DIMS + i * V4S) * DIMS + j * V4S);
            const unsigned long long gaW = (unsigned long long)(size_t)(const void*)
                (vw + (((size_t)s * V4D + i) * V4D + j) * K2);
            tdm4 gz; gz[0] = 0u; gz[1] = 0u; gz[2] = 0u; gz[3] = 0u;

            // ---- A: 3D tile (8 x 8 x 64 batches) ----
            tdm4 a0;
            a0[0] = 1u;
            a0[1] = lda;
            a0[2] = (unsigned)gaA;
            a0[3] = (unsigned)((gaA >> 32) & 0x01FFFFFFull) | (2u << 30);
            tdm8 a1;
            a1[0] = padctl;
            a1[1] = 8u << 16;             // tensor_dim0 = 8
            a1[2] = 8u << 16;             // tensor_dim1 = 8
            a1[3] = 8u << 16;             // tile_dim0   = 8
            a1[4] = 8u | (64u << 16);     // tile_dim1 = 8, tile_dim2 = 64
            a1[5] = (unsigned)DIMS;       // dim0_stride = 120 (rows of the map)
            {
                const unsigned long long s1 = (unsigned long long)SN * DIMS * DIMS; // 115200
                a1[6] = (unsigned)((s1 & 0xFFFFull) << 16);   // dim1_stride lo16
                a1[7] = (unsigned)(s1 >> 16);                 // dim1_stride hi32
            }
            tdm4 a2; a2[0] = 64u; a2[1] = 0u; a2[2] = 0u; a2[3] = 0u; // tensor_dim2
            tdm_load(a0, a1, a2, gz);

            // ---- W: 2D tile (64 k x 16 o) ----
            tdm4 w0;
            w0[0] = 1u;
            w0[1] = ldw;
            w0[2] = (unsigned)gaW;
            w0[3] = (unsigned)((gaW >> 32) & 0x01FFFFFFull) | (2u << 30);
            tdm8 w1;
            w1[0] = padctl;
            w1[1] = 64u << 16;                       // tensor_dim0 = 64
            w1[2] = 16u << 16;                       // tensor_dim1 = 16
            w1[3] = 64u << 16;                       // tile_dim0   = 64
            w1[4] = 16u;                             // tile_dim1   = 16
            w1[5] = (unsigned)(SN * V4D * V4D * K2); // dim0_stride (o-stride) = 430592
            w1[6] = 0u; w1[7] = 0u;
            tdm_load(w0, w1, gz, gz);

            __builtin_amdgcn_s_wait_tensorcnt(0);
        }
        __syncthreads();

#pragma unroll
        for (int kk = 0; kk < K2 / 4; ++kk) {                 // 16 steps
            v2f a = *(const v2f*)(ap + kk * 4);
            v2f b = *(const v2f*)(bp + kk * 4);
            acc = __builtin_amdgcn_wmma_f32_16x16x4_f32(
                false, a, false, b, (short)0, acc, false, false);
        }
    }

    // store v4[b, o, i, j] flattened as b*13456 + o*841 + i*29 + j
#pragma unroll
    for (int v = 0; v < 8; ++v) {
        const int b = mt * 16 + half * 8 + v;
        const int o = l16;
        v4[(size_t)b * V4FLAT + o * (V4D * V4D) + i * V4D + j] = acc[v];
    }
}

// =====================================================================
// Kernel 3: decision readout, [64 x 13456] @ [13456 x 2] + bias.
// One workgroup per batch element; fixed-order tree reduction (deterministic).
// =====================================================================
__global__ __launch_bounds__(256) void lcn_dec_kernel(
    const float* __restrict__ v4,     // [64, 13456]
    const float* __restrict__ dw,     // [2, 13456]
    const float* __restrict__ db,     // [2]
    float* __restrict__ out)          // [64, 2]
{
    __shared__ float red0[256];
    __shared__ float red1[256];
    const int b = blockIdx.x;
    const int tid = threadIdx.x;

    float p0 = 0.f, p1 = 0.f;
    for (int k = tid; k < V4FLAT; k += 256) {
        const float f = v4[(size_t)b * V4FLAT + k];
        p0 += f * dw[k];
        p1 += f * dw[V4FLAT + k];
    }
    red0[tid] = p0; red1[tid] = p1;
    __syncthreads();
    for (int st = 128; st > 0; st >>= 1) {
        if (tid < st) { red0[tid] += red0[tid + st]; red1[tid] += red1[tid + st]; }
        __syncthreads();
    }
    if (tid == 0) {
        out[b * 2 + 0] = red0[0] + db[0];
        out[b * 2 + 1] = red1[0] + db[1];
    }
}

// =====================================================================
extern "C" void kernel_launch(void* const* d_in, const int* in_sizes, int n_in,
                              void* d_out, int out_size, void* d_ws, size_t ws_size,
                              hipStream_t stream)
{
    (void)in_sizes; (void)n_in; (void)out_size; (void)ws_size;
    const float* x  = (const float*)d_in[0];   // [64,1,128,128]
    const float* sw = (const float*)d_in[1];   // [32,120,120,81]
    const float* vw = (const float*)d_in[2];   // [16,8,29,29,64]
    const float* dw = (const float*)d_in[3];   // [2,13456]
    const float* db = (const float*)d_in[4];   // [2]
    float* out = (float*)d_out;                // [64,2]

    // workspace layout: cplx [64*8*120*120] floats, then v4 [64*13456] floats
    float* cplx = (float*)d_ws;
    float* v4   = cplx + (size_t)BB * SN * DIMS * DIMS;

    lcn_v1_kernel<<<DIMS * DIMS, 256, 0, stream>>>(x, sw, cplx);
    lcn_v4_kernel<<<V4D * V4D, 128, 0, stream>>>(cplx, vw, v4);
    lcn_dec_kernel<<<BB, 256, 0, stream>>>(v4, dw, db, out);
}